// MAGC_6725918786010
// MI455X (gfx1250) — compile-verified
//
#include <hip/hip_runtime.h>
#include <math.h>

typedef __attribute__((ext_vector_type(2))) float v2f;
typedef __attribute__((ext_vector_type(8))) float v8f;

#define NB   32
#define NC   512
#define NG   8
#define ND   64
#define NP   4096        // 64*64 spatial positions
#define NPL  32          // planes
#define LN_EPS 1e-3f

// ---------------------------------------------------------------------------
// K1: per-(batch,head) attention logits -> softmax -> context vector.
// One workgroup per (b,g); 256 threads (8 waves, wave32).
// ---------------------------------------------------------------------------
__global__ __launch_bounds__(256) void magc_attn_ctx(
    const float* __restrict__ x, const float* __restrict__ w_mask,
    const float* __restrict__ b_mask, float* __restrict__ ctx)
{
    __shared__ float s_attn[NP];        // 16 KB: logits -> exp values
    __shared__ float s_red[16];
    __shared__ float s_ctx[16][ND];     // partial context sums

    const int bg  = blockIdx.x;         // 0..255
    const int b   = bg >> 3;
    const int g   = bg & 7;
    const int tid = threadIdx.x;
    const size_t base = (size_t)b * NP * NC + (size_t)g * ND;

    // ---- Pass 1: logits[p] = dot(x[p, g, 0:64], w_mask) + b_mask ----------
    // 16 lanes cooperate on one pixel: lane j loads float4 at d = 4j.
    const int j  = tid & 15;            // sub-lane within half-wave
    const int hw = tid >> 4;            // half-wave id 0..15
    const float4 wm = ((const float4*)w_mask)[j];
    const float  bm = b_mask[0];

    for (int p0 = 0; p0 < NP; p0 += 16) {
        const int p = p0 + hw;
        const float4 xv = ((const float4*)(x + base + (size_t)p * NC))[j];
        float dot = xv.x*wm.x + xv.y*wm.y + xv.z*wm.z + xv.w*wm.w;
        #pragma unroll
        for (int off = 8; off >= 1; off >>= 1)
            dot += __shfl_xor(dot, off, 32);   // reduce within 16-lane group
        if (j == 0) s_attn[p] = dot + bm;
    }
    __syncthreads();

    // ---- Softmax: block max ------------------------------------------------
    float mx = -INFINITY;
    for (int p = tid; p < NP; p += 256) mx = fmaxf(mx, s_attn[p]);
    #pragma unroll
    for (int off = 16; off >= 1; off >>= 1) mx = fmaxf(mx, __shfl_xor(mx, off, 32));
    if ((tid & 31) == 0) s_red[tid >> 5] = mx;
    __syncthreads();
    if (tid < 32) {
        float m2 = (tid < 8) ? s_red[tid] : -INFINITY;
        #pragma unroll
        for (int off = 4; off >= 1; off >>= 1) m2 = fmaxf(m2, __shfl_xor(m2, off, 32));
        if (tid == 0) s_red[0] = m2;
    }
    __syncthreads();
    mx = s_red[0];

    // ---- exp and block sum -------------------------------------------------
    float ssum = 0.f;
    for (int p = tid; p < NP; p += 256) {
        float e = __expf(s_attn[p] - mx);
        s_attn[p] = e;
        ssum += e;
    }
    #pragma unroll
    for (int off = 16; off >= 1; off >>= 1) ssum += __shfl_xor(ssum, off, 32);
    if ((tid & 31) == 0) s_red[8 + (tid >> 5)] = ssum;
    __syncthreads();
    if (tid < 32) {
        float s2 = (tid < 8) ? s_red[8 + tid] : 0.f;
        #pragma unroll
        for (int off = 4; off >= 1; off >>= 1) s2 += __shfl_xor(s2, off, 32);
        if (tid == 0) s_red[0] = s2;
    }
    __syncthreads();
    const float inv = 1.0f / s_red[0];  // fold 1/sum into ctx at the end

    // ---- Pass 2: ctx[d] = sum_p attn[p] * x[p, g, d] -----------------------
    // lane layout: d4 = tid&15 (float4 along d, coalesced), chunk = tid>>4
    const int d4    = tid & 15;
    const int chunk = tid >> 4;
    float4 acc = make_float4(0.f, 0.f, 0.f, 0.f);
    const int pbeg = chunk * (NP / 16);
    #pragma unroll 4
    for (int p = pbeg; p < pbeg + NP / 16; ++p) {
        const float4 xv = ((const float4*)(x + base + (size_t)p * NC))[d4];
        const float a = s_attn[p];
        acc.x += a * xv.x; acc.y += a * xv.y; acc.z += a * xv.z; acc.w += a * xv.w;
    }
    s_ctx[chunk][d4*4+0] = acc.x;
    s_ctx[chunk][d4*4+1] = acc.y;
    s_ctx[chunk][d4*4+2] = acc.z;
    s_ctx[chunk][d4*4+3] = acc.w;
    __syncthreads();
    if (tid < ND) {
        float v = 0.f;
        #pragma unroll
        for (int c = 0; c < 16; ++c) v += s_ctx[c][tid];
        ctx[(size_t)b * NC + g * ND + tid] = v * inv;
    }
}

// ---------------------------------------------------------------------------
// K2: tiny MLP on V_WMMA_F32_16X16X4_F32.
//   t = ctx(32x512) @ w1(512x32) + b1 ; LayerNorm(32) ; ReLU
//   out = t(32x32)  @ w2(32x512)  + b2
// Single workgroup, 4 waves. VGPR fragment layouts per CDNA5 ISA 7.12.2:
//   A 16x4 f32:  lanes 0-15 hold M=0..15; VGPR0/1 = K0/K1 (low half), K2/K3 (high)
//   B 4x16 f32:  lanes 0-15 hold N=0..15; VGPR0/1 = K0/K1 (low half), K2/K3 (high)
//   D 16x16 f32: VGPR r -> M=r (lanes 0-15), M=r+8 (lanes 16-31), N=lane&15
// ---------------------------------------------------------------------------
__global__ __launch_bounds__(128) void magc_mlp(
    const float* __restrict__ ctx, const float* __restrict__ w1,
    const float* __restrict__ b1,  const float* __restrict__ gamma,
    const float* __restrict__ beta, const float* __restrict__ w2,
    const float* __restrict__ b2,  float* __restrict__ out)
{
    __shared__ float s_t[NB][NPL + 1];  // t matrix, padded

    const int tid  = threadIdx.x;
    const int wave = tid >> 5;
    const int lane = tid & 31;
    const int lo   = lane & 15;
    const int hiK  = (lane >> 4) << 1;  // K offset: 0 (low half) / 2 (high half)
    const int hi8  = (lane >> 4) << 3;  // M offset for D rows: 0 / 8

    // ---- GEMM1: each wave owns one 16x16 tile of t (2x2 tiles) -------------
    {
        const int m16 = (wave >> 1) & 1;
        const int n16 = wave & 1;
        v8f acc = {};
        for (int k0 = 0; k0 < NC; k0 += 4) {
            v2f a, bf;
            const float* arow = ctx + (size_t)(m16*16 + lo) * NC + k0 + hiK;
            a.x = arow[0];
            a.y = arow[1];
            const float* bcol = w1 + (size_t)(k0 + hiK) * NPL + n16*16 + lo;
            bf.x = bcol[0];
            bf.y = bcol[NPL];
            acc = __builtin_amdgcn_wmma_f32_16x16x4_f32(
                false, a, false, bf, (short)0, acc, false, false);
        }
        const int N = n16*16 + lo;
        const float bb = b1[N];
        #pragma unroll
        for (int r = 0; r < 8; ++r)
            s_t[m16*16 + hi8 + r][N] = acc[r] + bb;
    }
    __syncthreads();

    // ---- LayerNorm over 32 planes + ReLU, one thread per batch row ---------
    if (tid < NB) {
        float mu = 0.f;
        #pragma unroll
        for (int p = 0; p < NPL; ++p) mu += s_t[tid][p];
        mu *= (1.0f / NPL);
        float var = 0.f;
        #pragma unroll
        for (int p = 0; p < NPL; ++p) { float dv = s_t[tid][p] - mu; var += dv*dv; }
        var *= (1.0f / NPL);
        const float rinv = rsqrtf(var + LN_EPS);
        #pragma unroll
        for (int p = 0; p < NPL; ++p) {
            float v = gamma[p] * (s_t[tid][p] - mu) * rinv + beta[p];
            s_t[tid][p] = fmaxf(v, 0.f);
        }
    }
    __syncthreads();

    // ---- GEMM2: 2 M-tiles x 32 N-tiles = 64 tiles over 4 waves -------------
    for (int tile = wave; tile < 64; tile += 4) {
        const int mt = tile & 1;
        const int nt = tile >> 1;
        v8f acc = {};
        #pragma unroll
        for (int k0 = 0; k0 < NPL; k0 += 4) {
            v2f a, bf;
            a.x = s_t[mt*16 + lo][k0 + hiK + 0];
            a.y = s_t[mt*16 + lo][k0 + hiK + 1];
            const float* bcol = w2 + (size_t)(k0 + hiK) * NC + nt*16 + lo;
            bf.x = bcol[0];
            bf.y = bcol[NC];
            acc = __builtin_amdgcn_wmma_f32_16x16x4_f32(
                false, a, false, bf, (short)0, acc, false, false);
        }
        const int N = nt*16 + lo;
        const float bb = b2[N];
        #pragma unroll
        for (int r = 0; r < 8; ++r)
            out[(size_t)(mt*16 + hi8 + r) * NC + N] = acc[r] + bb;
    }
}

// ---------------------------------------------------------------------------
// K3: y = x + out[b, c] broadcast. float4 coalesced; out (64 KB) stays in L2.
// ---------------------------------------------------------------------------
__global__ __launch_bounds__(256) void magc_add(
    const float* __restrict__ x, const float* __restrict__ mlp_out,
    float* __restrict__ y)
{
    const long long i = (long long)blockIdx.x * 256 + threadIdx.x;  // float4 idx
    const float4 xv = ((const float4*)x)[i];
    const int c4 = (int)(i & (NC/4 - 1));   // 128 float4 per pixel
    const int b  = (int)(i >> 19);          // / (4096 * 128)
    const float4 ov = ((const float4*)mlp_out)[(size_t)b * (NC/4) + c4];
    float4 r;
    r.x = xv.x + ov.x; r.y = xv.y + ov.y; r.z = xv.z + ov.z; r.w = xv.w + ov.w;
    ((float4*)y)[i] = r;
}

extern "C" void kernel_launch(void* const* d_in, const int* in_sizes, int n_in,
                              void* d_out, int out_size, void* d_ws, size_t ws_size,
                              hipStream_t stream) {
    const float* x      = (const float*)d_in[0];
    const float* w_mask = (const float*)d_in[1];
    const float* b_mask = (const float*)d_in[2];
    const float* w1     = (const float*)d_in[3];
    const float* b1     = (const float*)d_in[4];
    const float* gamma  = (const float*)d_in[5];
    const float* beta   = (const float*)d_in[6];
    const float* w2     = (const float*)d_in[7];
    const float* b2     = (const float*)d_in[8];
    float* y = (float*)d_out;

    float* ctx = (float*)d_ws;        // 32*512 f32 = 64 KB
    float* mlp = ctx + NB * NC;       // 32*512 f32 = 64 KB

    magc_attn_ctx<<<NB * NG, 256, 0, stream>>>(x, w_mask, b_mask, ctx);
    magc_mlp<<<1, 128, 0, stream>>>(ctx, w1, b1, gamma, beta, w2, b2, mlp);
    magc_add<<<(NB * NP * (NC / 4)) / 256, 256, 0, stream>>>(x, mlp, y);
}